// GlobalProdRenormNormalConstCovStdPrior_38439957299402
// MI455X (gfx1250) — compile-verified
//
#include <hip/hip_runtime.h>
#include <hip/hip_bf16.h>

// ---------------------------------------------------------------------------
// GlobalProdRenormNormalConstCovStdPrior for MI455X (gfx1250)
// B=64, T=2048, U=512.
// Heavy parts: 64x batched 512x512 SPD inversion + Cholesky, done as blocked
// tile algorithms driven by V_WMMA_F32_16X16X4_F32 (fp32 matrix cores).
// Matrices live in the d_out chol region (64 MB -> resident in 192MB L2).
// ---------------------------------------------------------------------------

#define NTILE 32            // 512 / 16 tiles per dim
#define TS 17               // padded LDS tile row stride (bank-conflict free)
#define TILE_FLOATS (16*TS)

typedef __attribute__((ext_vector_type(2))) float v2f;
typedef __attribute__((ext_vector_type(8))) float v8f;

// f32 WMMA: ISA says NEG[1:0] must be 0 for F32 (only C-negate exists), so
// pass neg_a=neg_b=false and fold any sign into the fragments.
#define WMMA_F32(a, b, c) \
  __builtin_amdgcn_wmma_f32_16x16x4_f32(false, (a), false, (b), (short)0, (c), false, false)

// ---------------------------------------------------------------------------
// K1: zero init for atomics
__global__ __launch_bounds__(512) void k_zero(float* __restrict__ p, int n) {
  int i = blockIdx.x * 512 + threadIdx.x;
  if (i < n) p[i] = 0.f;
}

// ---------------------------------------------------------------------------
// K2: gamma[b][u] = sum_t x[b][t][u]*w[b][t]; N[b] = sum_t w[b][t]
// grid (64, 16), block 512. Reads the 268 MB x tensor once, coalesced.
__global__ __launch_bounds__(512) void k_reduce(const float* __restrict__ x,
                                                const float* __restrict__ w,
                                                float* __restrict__ gamma,
                                                float* __restrict__ nsum) {
  const int b = blockIdx.x;
  const int t0 = blockIdx.y * 128;
  const int u = threadIdx.x;
  const float* xb = x + ((long long)b * 2048 + t0) * 512 + u;
  const float* wb = w + b * 2048 + t0;
  float acc = 0.f, wsum = 0.f;
  for (int t = 0; t < 128; ++t) {
    float wt = wb[t];
    acc = fmaf(xb[(long long)t * 512], wt, acc);
    wsum += wt;
  }
  atomicAdd(&gamma[b * 512 + u], acc);
  if (u == 0) atomicAdd(&nsum[b], wsum);
}

// ---------------------------------------------------------------------------
// K3: S[k][j] = exp(0.5*D[k]) * chol[k][j]  (so cov_i = S^T S)
__global__ __launch_bounds__(512) void k_scaleS(const float* __restrict__ D,
                                                const float* __restrict__ chol,
                                                float* __restrict__ S) {
  int k = blockIdx.x, j = threadIdx.x;
  S[k * 512 + j] = expf(0.5f * D[k]) * chol[k * 512 + j];
}

// ---------------------------------------------------------------------------
// K4: cov_i = S^T S via WMMA f32 16x16x4. grid 1024 (one 16x16 tile / wave).
__global__ __launch_bounds__(32) void k_syrk_covi(const float* __restrict__ S,
                                                  float* __restrict__ C) {
  const int ti = blockIdx.x >> 5;
  const int tj = blockIdx.x & 31;
  const int lane = threadIdx.x;
  const int hf = lane >> 4;
  const int m  = lane & 15;
  v8f c = {};
  for (int k0 = 0; k0 < 512; k0 += 4) {
    int ka = k0 + hf * 2;
    v2f a, b;
    a.x = S[(long long)ka * 512 + ti * 16 + m];        // A[m][ka] = S[ka][ti*16+m]
    a.y = S[(long long)(ka + 1) * 512 + ti * 16 + m];
    b.x = S[(long long)ka * 512 + tj * 16 + m];        // B[ka][m] = S[ka][tj*16+m]
    b.y = S[(long long)(ka + 1) * 512 + tj * 16 + m];
    c = WMMA_F32(a, b, c);
  }
  float* g = C + (ti * 16) * 512 + tj * 16;
  for (int r = 0; r < 8; ++r) g[(r + hf * 8) * 512 + m] = c[r];
}

// ---------------------------------------------------------------------------
// K5: blocked in-place Gauss-Jordan inversion of a 512x512 matrix.
// One workgroup (512 thr = 16 waves) per matrix; 32 block-pivot steps.
// Trailing updates are WMMA 16x16x16 tile-GEMMs. Column panel stored negated
// so trailing is C += (-col)*row (f32 WMMA has no A-negate).
__global__ __launch_bounds__(512) void k_gj_inv(float* __restrict__ mats,
                                                long long mstride) {
  float* A = mats + (long long)blockIdx.x * mstride;
  __shared__ float piv[16 * TS];
  __shared__ float rowP[NTILE * TILE_FLOATS];
  __shared__ float colP[NTILE * TILE_FLOATS];
  const int tid = threadIdx.x;
  const int lane = tid & 31;
  const int wv = tid >> 5;       // 0..15
  const int hf = lane >> 4;
  const int m  = lane & 15;
  const int ei = tid >> 4;       // (tid<256) element coords
  const int ej = tid & 15;

  for (int kb = 0; kb < NTILE; ++kb) {
    // --- Phase A: load + invert pivot tile (scalar in-place GJ, 16 steps) ---
    if (tid < 256) piv[ei * TS + ej] = A[(kb * 16 + ei) * 512 + kb * 16 + ej];
    __syncthreads();
    for (int k = 0; k < 16; ++k) {
      float p = 0.f, aik = 0.f, akj = 0.f, aij = 0.f;
      if (tid < 256) {
        p = 1.0f / piv[k * TS + k];
        aik = piv[ei * TS + k];
        akj = piv[k * TS + ej];
        aij = piv[ei * TS + ej];
      }
      __syncthreads();
      if (tid < 256) {
        float r;
        if (ei == k)      r = (ej == k) ? p : akj * p;
        else if (ej == k) r = -aik * p;
        else              r = fmaf(-aik * p, akj, aij);
        piv[ei * TS + ej] = r;
      }
      __syncthreads();
    }
    // --- Phase A2: stage old row panel (raw) and old col panel (negated) ---
    for (int idx = tid; idx < NTILE * 256; idx += 512) {
      int t = idx >> 8, e = idx & 255;
      int i = e >> 4, j = e & 15;
      rowP[t * TILE_FLOATS + i * TS + j] =  A[(kb * 16 + i) * 512 + t * 16 + j];
      colP[t * TILE_FLOATS + i * TS + j] = -A[(t * 16 + i) * 512 + kb * 16 + j];
    }
    __syncthreads();
    // --- Phase B: newRow = Pinv*oldRow ; newCol = (-oldCol)*Pinv ; A[kb][kb]=Pinv
    for (int t = wv; t < NTILE; t += 16) {           // wave-uniform loop
      if (t == kb) {
        for (int r = 0; r < 8; ++r) {
          int row = r + hf * 8;
          A[(kb * 16 + row) * 512 + kb * 16 + m] = piv[row * TS + m];
        }
        continue;
      }
      v8f cr = {}, cc = {};
      for (int k0 = 0; k0 < 16; k0 += 4) {
        int ka = k0 + hf * 2;
        v2f a, b;
        a.x = piv[m * TS + ka];  a.y = piv[m * TS + ka + 1];
        b.x = rowP[t * TILE_FLOATS + ka * TS + m];
        b.y = rowP[t * TILE_FLOATS + (ka + 1) * TS + m];
        cr = WMMA_F32(a, b, cr);
        v2f a2, b2;
        a2.x = colP[t * TILE_FLOATS + m * TS + ka];
        a2.y = colP[t * TILE_FLOATS + m * TS + ka + 1];
        b2.x = piv[ka * TS + m];  b2.y = piv[(ka + 1) * TS + m];
        cc = WMMA_F32(a2, b2, cc);
      }
      for (int r = 0; r < 8; ++r) {
        int row = r + hf * 8;
        rowP[t * TILE_FLOATS + row * TS + m] = cr[r];      // keep newRow for trailing
        A[(kb * 16 + row) * 512 + t * 16 + m] = cr[r];
        A[(t * 16 + row) * 512 + kb * 16 + m] = cc[r];
      }
    }
    __syncthreads();
    // --- Phase C: trailing  A[i][j] += (-oldCol_i) * newRow_j  (i,j != kb) ---
    for (int idx = wv; idx < NTILE * NTILE; idx += 16) {   // wave-uniform
      int ti = idx >> 5, tj = idx & 31;
      if (ti == kb || tj == kb) continue;
      float* g = A + (ti * 16) * 512 + tj * 16;
      v8f c;
      for (int r = 0; r < 8; ++r) c[r] = g[(r + hf * 8) * 512 + m];
      for (int k0 = 0; k0 < 16; k0 += 4) {
        int ka = k0 + hf * 2;
        v2f a, b;
        a.x = colP[ti * TILE_FLOATS + m * TS + ka];
        a.y = colP[ti * TILE_FLOATS + m * TS + ka + 1];
        b.x = rowP[tj * TILE_FLOATS + ka * TS + m];
        b.y = rowP[tj * TILE_FLOATS + (ka + 1) * TS + m];
        c = WMMA_F32(a, b, c);
      }
      for (int r = 0; r < 8; ++r) g[(r + hf * 8) * 512 + m] = c[r];
    }
    __threadfence();
    __syncthreads();
  }
}

// ---------------------------------------------------------------------------
// K6: prec_b = I + N_b * prec_i  written into the per-batch matrix region
__global__ __launch_bounds__(512) void k_form_prec(const float* __restrict__ preci,
                                                   const float* __restrict__ nsum,
                                                   float* __restrict__ cov) {
  int b = blockIdx.x, i = blockIdx.y, j = threadIdx.x;
  float nb = nsum[b];
  cov[((long long)b * 512 + i) * 512 + j] =
      ((i == j) ? 1.f : 0.f) + nb * preci[i * 512 + j];
}

// ---------------------------------------------------------------------------
// K7: cov = 0.5*(cov + cov^T)  (pair-owned, race-free)
__global__ __launch_bounds__(512) void k_symm(float* __restrict__ cov) {
  int b = blockIdx.x, i = blockIdx.y, j = threadIdx.x;
  if (j > i) {
    float* M = cov + (long long)b * 512 * 512;
    float av = 0.5f * (M[i * 512 + j] + M[j * 512 + i]);
    M[i * 512 + j] = av;
    M[j * 512 + i] = av;
  }
}

// ---------------------------------------------------------------------------
// K8: mu[b][v] = sum_u gamma[b][u] * cov[b][u][v]
__global__ __launch_bounds__(512) void k_mu(const float* __restrict__ gamma,
                                            const float* __restrict__ cov,
                                            float* __restrict__ mu) {
  int b = blockIdx.x, v = threadIdx.x;
  const float* M = cov + (long long)b * 512 * 512;
  const float* g = gamma + b * 512;
  float acc = 0.f;
  for (int u = 0; u < 512; ++u) acc = fmaf(g[u], M[u * 512 + v], acc);
  mu[b * 512 + v] = acc;
}

// ---------------------------------------------------------------------------
// K9: blocked right-looking Cholesky (lower) in place, WMMA trailing SYRK.
__global__ __launch_bounds__(512) void k_chol(float* __restrict__ mats,
                                              long long mstride) {
  float* A = mats + (long long)blockIdx.x * mstride;
  __shared__ float dt[16 * TS];      // diag tile -> L_kk
  __shared__ float linv[16 * TS];    // L_kk^{-1} (lower)
  __shared__ float panel[NTILE * TILE_FLOATS];
  const int tid = threadIdx.x;
  const int lane = tid & 31;
  const int wv = tid >> 5;
  const int hf = lane >> 4;
  const int m  = lane & 15;
  const int ei = tid >> 4, ej = tid & 15;

  for (int kb = 0; kb < NTILE; ++kb) {
    if (tid < 256) dt[ei * TS + ej] = A[(kb * 16 + ei) * 512 + kb * 16 + ej];
    __syncthreads();
    // 16x16 Cholesky (lower), sequential over k
    for (int k = 0; k < 16; ++k) {
      if (tid == 0) dt[k * TS + k] = sqrtf(dt[k * TS + k]);
      __syncthreads();
      if (tid < 16 && tid > k) dt[tid * TS + k] /= dt[k * TS + k];
      __syncthreads();
      if (tid < 256 && ei > k && ej > k && ej <= ei)
        dt[ei * TS + ej] = fmaf(-dt[ei * TS + k], dt[ej * TS + k], dt[ei * TS + ej]);
      __syncthreads();
    }
    // linv = inverse of lower L_kk (forward substitution, column per thread)
    for (int r = 0; r < 16; ++r) {
      if (tid < 16) {
        int c = tid;
        float s = (r == c) ? 1.f : 0.f;
        for (int t2 = 0; t2 < r; ++t2) s = fmaf(-dt[r * TS + t2], linv[t2 * TS + c], s);
        linv[r * TS + c] = s / dt[r * TS + r];
      }
      __syncthreads();
    }
    // store L_kk with zero upper
    if (tid < 256)
      A[(kb * 16 + ei) * 512 + kb * 16 + ej] = (ej <= ei) ? dt[ei * TS + ej] : 0.f;
    __syncthreads();
    // panel: for i>kb : P_i = A[i][kb] * linv^T  (WMMA)
    for (int t = kb + 1 + wv; t < NTILE; t += 16) {       // wave-uniform
      v8f c = {};
      for (int k0 = 0; k0 < 16; k0 += 4) {
        int ka = k0 + hf * 2;
        v2f a, b;
        a.x = A[(t * 16 + m) * 512 + kb * 16 + ka];
        a.y = A[(t * 16 + m) * 512 + kb * 16 + ka + 1];
        b.x = linv[m * TS + ka];          // (linv^T)[ka][m] = linv[m][ka]
        b.y = linv[m * TS + ka + 1];
        c = WMMA_F32(a, b, c);
      }
      for (int r = 0; r < 8; ++r) {
        int row = r + hf * 8;
        panel[t * TILE_FLOATS + row * TS + m] = c[r];
        A[(t * 16 + row) * 512 + kb * 16 + m] = c[r];
      }
    }
    __syncthreads();
    // trailing SYRK: i>=j>kb : A[i][j] -= P_i * P_j^T  (WMMA, A-side negated)
    for (int idx = wv; idx < NTILE * NTILE; idx += 16) {  // wave-uniform
      int ti = idx >> 5, tj = idx & 31;
      if (tj <= kb || ti < tj) continue;
      float* g = A + (ti * 16) * 512 + tj * 16;
      v8f c;
      for (int r = 0; r < 8; ++r) c[r] = g[(r + hf * 8) * 512 + m];
      for (int k0 = 0; k0 < 16; k0 += 4) {
        int ka = k0 + hf * 2;
        v2f a, b;
        a.x = -panel[ti * TILE_FLOATS + m * TS + ka];
        a.y = -panel[ti * TILE_FLOATS + m * TS + ka + 1];
        b.x = panel[tj * TILE_FLOATS + m * TS + ka];      // (P_j^T)[ka][m]
        b.y = panel[tj * TILE_FLOATS + m * TS + ka + 1];
        c = WMMA_F32(a, b, c);
      }
      for (int r = 0; r < 8; ++r) g[(r + hf * 8) * 512 + m] = c[r];
    }
    __threadfence();
    __syncthreads();
  }
}

// ---------------------------------------------------------------------------
// K10: sigma = diag(L); logvar = 2*log(sigma)
__global__ __launch_bounds__(512) void k_fin_diag(const float* __restrict__ cov,
                                                  float* __restrict__ sigma,
                                                  float* __restrict__ logvar) {
  int b = blockIdx.x, i = threadIdx.x;
  float s = cov[((long long)b * 512 + i) * 512 + i];
  sigma[b * 512 + i] = s;
  logvar[b * 512 + i] = 2.0f * logf(s);
}

// K11: chol_out = L^T with rows normalized by diag; in place, pair-owned.
__global__ __launch_bounds__(512) void k_fin_off(float* __restrict__ cov,
                                                 const float* __restrict__ sigma) {
  int b = blockIdx.x, i = blockIdx.y, j = threadIdx.x;
  float* M = cov + (long long)b * 512 * 512;
  if (j > i) {
    float v = M[j * 512 + i];           // L[j][i]  (lower)
    M[i * 512 + j] = v / sigma[b * 512 + i];
    M[j * 512 + i] = 0.f;
  } else if (j == i) {
    M[i * 512 + i] = 1.f;
  }
}

// ---------------------------------------------------------------------------
extern "C" void kernel_launch(void* const* d_in, const int* in_sizes, int n_in,
                              void* d_out, int out_size, void* d_ws, size_t ws_size,
                              hipStream_t stream) {
  (void)in_sizes; (void)n_in; (void)out_size; (void)ws_size;
  const float* x    = (const float*)d_in[0];   // [64,2048,512]
  const float* w    = (const float*)d_in[1];   // [64,2048,1]
  const float* D    = (const float*)d_in[2];   // [512]
  const float* chol = (const float*)d_in[3];   // [512,512]

  float* out    = (float*)d_out;
  float* mu     = out;                  // [64,512]
  float* logvar = out + 64 * 512;       // [64,512]
  float* cov    = out + 2 * 64 * 512;   // [64,512,512] workspace -> chol_out

  float* ws    = (float*)d_ws;
  float* gamma = ws;                    // 32768
  float* nsum  = ws + 32768;            // 64
  float* sigma = ws + 32832;            // 32768
  float* S     = ws + 65600;            // 262144
  float* covi  = ws + 65600 + 262144;   // 262144  (~2.36 MB total)

  k_zero     <<<(32832 + 511) / 512, 512, 0, stream>>>(gamma, 32832);
  k_reduce   <<<dim3(64, 16), 512, 0, stream>>>(x, w, gamma, nsum);
  k_scaleS   <<<512, 512, 0, stream>>>(D, chol, S);
  k_syrk_covi<<<1024, 32, 0, stream>>>(S, covi);
  k_gj_inv   <<<1, 512, 0, stream>>>(covi, 0);                 // covi -> prec_i
  k_form_prec<<<dim3(64, 512), 512, 0, stream>>>(covi, nsum, cov);
  k_gj_inv   <<<64, 512, 0, stream>>>(cov, (long long)512 * 512); // -> cov_b
  k_symm     <<<dim3(64, 512), 512, 0, stream>>>(cov);
  k_mu       <<<64, 512, 0, stream>>>(gamma, cov, mu);
  k_chol     <<<64, 512, 0, stream>>>(cov, (long long)512 * 512); // -> L (lower)
  k_fin_diag <<<64, 512, 0, stream>>>(cov, sigma, logvar);
  k_fin_off  <<<dim3(64, 512), 512, 0, stream>>>(cov, sigma);
}